// UnsupervisedGIN_9174050144734
// MI455X (gfx1250) — compile-verified
//
#include <hip/hip_runtime.h>
#include <cstdint>

#define N_NODES 50000
#define N_EDGES 800000
#define D 64
#define NEG_SLOPE 0.01f

typedef float v2f __attribute__((ext_vector_type(2)));
typedef float v8f __attribute__((ext_vector_type(8)));

// Order-preserving float -> uint key (total order matching float compare for
// finite values; -inf maps to the minimum-ish key). Lets us use native
// global_atomic_max_u32 instead of a CAS loop for segment_max.
__device__ __forceinline__ uint32_t fkey(float f) {
  uint32_t u = __float_as_uint(f);
  return (u & 0x80000000u) ? ~u : (u | 0x80000000u);
}

// fkey(-inf): bits(-inf)=0xFF800000, sign set -> ~u = 0x007FFFFF
#define INIT_KEY 0x007FFFFFu

__global__ void init_keys_kernel(uint32_t* __restrict__ keys, int n) {
  int i = blockIdx.x * blockDim.x + threadIdx.x;
  if (i < n) keys[i] = INIT_KEY;
}

// One thread per (edge, feature). 64 consecutive threads share one edge ->
// fully coalesced 256B row reads of h[src]; atomic u32 max scatter to dst row.
__global__ void scatter_max_kernel(const float* __restrict__ h,
                                   const int* __restrict__ src,
                                   const int* __restrict__ dst,
                                   uint32_t* __restrict__ keys) {
  long long tid = (long long)blockIdx.x * blockDim.x + threadIdx.x;
  int e = (int)(tid >> 6);
  int f = (int)(tid & 63);
  if (e >= N_EDGES) return;
  int s = src[e];
  int d = dst[e];
  float v = h[(long long)s * D + f];
  atomicMax(&keys[(long long)d * D + f], fkey(v));
}

// Fused: x = (1+eps)*h + decode(agg_keys); out = x @ W + b (+ leaky relu).
// Block = 128 threads = 4 waves; block owns 16 rows, wave w owns cols [16w,16w+16).
// K=64 accumulated via 16 x V_WMMA_F32_16X16X4_F32.
__global__ void __launch_bounds__(128)
gin_update_kernel(const float* __restrict__ h,
                  const uint32_t* __restrict__ keys,
                  const float* __restrict__ W,
                  const float* __restrict__ bias,
                  const float* __restrict__ eps, int li,
                  float* __restrict__ out, int activate) {
  __shared__ float Ws[D * D];      // 16 KB, row-major W[k][n]
  __shared__ float Xs[16 * 65];    // 16-row tile, pitch 65 to avoid bank conflicts

  const int t    = threadIdx.x;
  const int wave = t >> 5;
  const int lane = t & 31;
  const int r0   = blockIdx.x * 16;
  const float ep1 = 1.0f + eps[li];

  // Stage W: 4096 floats via float4, 128 threads x 8 iters.
  const float4* W4  = (const float4*)W;
  float4*       Ws4 = (float4*)Ws;
#pragma unroll
  for (int i = 0; i < 8; ++i) Ws4[t + i * 128] = W4[t + i * 128];

  // Build X tile: (1+eps)*h + agg, where agg = decode(key) with non-finite -> 0
  // (init key decodes to -inf, so empty segments naturally become 0, matching
  //  the reference's isfinite() fill).
#pragma unroll
  for (int i = 0; i < 8; ++i) {
    int idx = t + i * 128;              // 0..1023
    int row = idx >> 6;
    int f   = idx & 63;
    long long g = (long long)(r0 + row) * D + f;
    uint32_t k = keys[g];
    uint32_t u = (k & 0x80000000u) ? (k & 0x7FFFFFFFu) : ~k;
    float av = __uint_as_float(u);
    float a  = (((u >> 23) & 0xFFu) != 0xFFu) ? av : 0.0f;  // isfinite
    Xs[row * 65 + f] = ep1 * h[g] + a;
  }
  __syncthreads();

  const int half = lane >> 4;   // 0: lanes 0-15, 1: lanes 16-31
  const int l16  = lane & 15;
  const int col  = wave * 16 + l16;

  v8f acc = {};
#pragma unroll
  for (int kk = 0; kk < 16; ++kk) {
    const int kbase = kk * 4 + half * 2;   // A layout: K=0,1 | K=2,3 per half
    v2f a, b;
    a.x = Xs[l16 * 65 + kbase + 0];
    a.y = Xs[l16 * 65 + kbase + 1];
    b.x = Ws[(kbase + 0) * D + col];       // B layout mirrors A: K rows, N=lane
    b.y = Ws[(kbase + 1) * D + col];
    acc = __builtin_amdgcn_wmma_f32_16x16x4_f32(
        /*neg_a=*/false, a, /*neg_b=*/false, b,
        /*c_mod=*/(short)0, acc, /*reuse_a=*/false, /*reuse_b=*/false);
  }

  const float bc = bias[col];
#pragma unroll
  for (int v = 0; v < 8; ++v) {
    int row = v + half * 8;                // C/D layout: VGPR v -> M=v / v+8
    float val = acc[v] + bc;
    if (activate) val = (val >= 0.0f) ? val : NEG_SLOPE * val;
    out[(long long)(r0 + row) * D + col] = val;
  }
}

extern "C" void kernel_launch(void* const* d_in, const int* in_sizes, int n_in,
                              void* d_out, int out_size, void* d_ws, size_t ws_size,
                              hipStream_t stream) {
  const float* n_feat = (const float*)d_in[0];
  const float* W0     = (const float*)d_in[1];
  const float* b0     = (const float*)d_in[2];
  const float* W1     = (const float*)d_in[3];
  const float* b1     = (const float*)d_in[4];
  const float* W2     = (const float*)d_in[5];
  const float* b2     = (const float*)d_in[6];
  const float* eps    = (const float*)d_in[7];
  const int*   src    = (const int*)d_in[8];
  const int*   dst    = (const int*)d_in[9];
  float*       out    = (float*)d_out;

  const size_t nd = (size_t)N_NODES * D;           // 3.2M elements
  uint32_t* keys = (uint32_t*)d_ws;                // 12.8 MB
  float*    h1   = (float*)((char*)d_ws + nd * 4); // 12.8 MB
  float*    h2   = (float*)((char*)d_ws + 2 * nd * 4);

  const float* hin[3]  = {n_feat, h1, h2};
  float*       hout[3] = {h1, h2, out};
  const float* Wl[3]   = {W0, W1, W2};
  const float* bl[3]   = {b0, b1, b2};

  const int nk = (int)nd;
  dim3 initGrid((nk + 255) / 256);
  dim3 scatGrid((unsigned)(((long long)N_EDGES * 64) / 256)); // 200000 blocks
  dim3 gemmGrid(N_NODES / 16);                                 // 3125 blocks

  for (int l = 0; l < 3; ++l) {
    init_keys_kernel<<<initGrid, 256, 0, stream>>>(keys, nk);
    scatter_max_kernel<<<scatGrid, 256, 0, stream>>>(hin[l], src, dst, keys);
    gin_update_kernel<<<gemmGrid, 128, 0, stream>>>(hin[l], keys, Wl[l], bl[l],
                                                    eps, l, hout[l],
                                                    (l < 2) ? 1 : 0);
  }
}